// HO_FGRU_AUG_70136815944401
// MI455X (gfx1250) — compile-verified
//
#include <hip/hip_runtime.h>
#include <math.h>

// ---------------------------------------------------------------------------
// Problem constants (from reference)
// ---------------------------------------------------------------------------
#define H_      768
#define L_      2
#define GS_     4
#define GROUPS_ 8
#define A_      4
#define F_      4097
#define M_      2
#define K_      3
#define P_      8
#define OUT_    8           // A*M
#define LAM_    0.01f
#define PAD_    3           // GS/2 + 1
#define TP_     4100        // (F + 2*PAD - GS)/HOP + 1
#define CIN_    32          // A*M*K + P
#define XROWS_  (F_ + 2*PAD_)   // 4103
#define H3_     (3*H_)      // 2304

// CDNA5 async-LDS staging path (global_load_async_to_lds_b128, ASYNCcnt).
#define USE_ASYNC_LDS 1

typedef float v2f __attribute__((ext_vector_type(2)));
typedef float v8f __attribute__((ext_vector_type(8)));

__device__ __forceinline__ float2 cmul(float2 a, float2 b) {
    return make_float2(a.x*b.x - a.y*b.y, a.x*b.y + a.y*b.x);
}
__device__ __forceinline__ float sigm(float x) { return 1.0f / (1.0f + expf(-x)); }

__device__ __forceinline__ v8f wmma_f32(v2f a, v2f b, v8f c) {
    return __builtin_amdgcn_wmma_f32_16x16x4_f32(false, a, false, b, (short)0, c, false, false);
}

__device__ __forceinline__ void wait_async_all() {
#if USE_ASYNC_LDS
#if __has_builtin(__builtin_amdgcn_s_wait_asynccnt)
    __builtin_amdgcn_s_wait_asynccnt(0);
#else
    asm volatile("s_wait_asynccnt 0" ::: "memory");
#endif
#endif
}

// ---------------------------------------------------------------------------
// 1) Pack full_inputs+u -> padded (4103, 32) with log1p-magnitude compression
// ---------------------------------------------------------------------------
__global__ void prep_kernel(const float2* __restrict__ fi, const float2* __restrict__ u,
                            float2* __restrict__ xpad) {
    int tid = blockIdx.x * blockDim.x + threadIdx.x;
    if (tid >= XROWS_ * CIN_) return;
    int row = tid / CIN_, c = tid % CIN_;
    float2 z = make_float2(0.f, 0.f);
    int f = row - PAD_;
    if (f >= 0 && f < F_) {
        float2 v;
        if (c < A_*M_*K_) {
            int a = c / (M_*K_);
            int m = (c / K_) % M_;
            int k = c % K_;
            v = fi[(((size_t)a*F_ + f)*M_ + m)*K_ + k];   // (A,F,M,K) row-major
        } else {
            v = u[(size_t)f*P_ + (c - A_*M_*K_)];
        }
        float mag = sqrtf(v.x*v.x + v.y*v.y);
        float s = (mag > 0.f) ? (log1pf(mag) / mag) : 0.f;  // log1p(|z|)*exp(i*angle)
        z.x = v.x * s; z.y = v.y * s;
    }
    xpad[tid] = z;
}

// ---------------------------------------------------------------------------
// 2) Complex conv1d (VALID, stride 1) + bias + CReLU : (4103,32) -> (4100,768)
// ---------------------------------------------------------------------------
__global__ void conv_kernel(const float2* __restrict__ xpad, const float2* __restrict__ w,
                            const float2* __restrict__ bias, float2* __restrict__ y) {
    int tid = blockIdx.x * blockDim.x + threadIdx.x;
    if (tid >= TP_ * H_) return;
    int t = tid / H_, h = tid % H_;
    float2 bv = bias[h];
    float ar = bv.x, ai = bv.y;
    for (int g = 0; g < GS_; ++g) {
        const float2* xr = xpad + (size_t)(t + g) * CIN_;
        const float2* wr = w + ((size_t)g * CIN_) * H_ + h;   // (GS,CIN,H) row-major
        #pragma unroll 4
        for (int c = 0; c < CIN_; ++c) {
            float2 xv = xr[c];
            float2 wv = wr[(size_t)c * H_];
            ar += xv.x*wv.x - xv.y*wv.y;
            ai += xv.x*wv.y + xv.y*wv.x;
        }
    }
    y[tid] = make_float2(fmaxf(ar, 0.f), fmaxf(ai, 0.f));    // crelu
}

// ---------------------------------------------------------------------------
// 3) Complex GroupNorm: stats (1 block per group), then apply
// ---------------------------------------------------------------------------
__global__ void gn_stats_kernel(const float2* __restrict__ x, float* __restrict__ stats) {
    const int g  = blockIdx.x;
    const int cs = H_ / GROUPS_;                  // 96 channels per group
    float sr = 0.f, si = 0.f, sq = 0.f;
    for (int i = threadIdx.x; i < TP_ * cs; i += blockDim.x) {
        int t = i / cs, c = g * cs + (i % cs);
        float2 v = x[(size_t)t * H_ + c];
        sr += v.x; si += v.y; sq += v.x*v.x + v.y*v.y;
    }
    __shared__ float shr[256], shi[256], shq[256];
    shr[threadIdx.x] = sr; shi[threadIdx.x] = si; shq[threadIdx.x] = sq;
    __syncthreads();
    for (int s = 128; s > 0; s >>= 1) {
        if ((int)threadIdx.x < s) {
            shr[threadIdx.x] += shr[threadIdx.x + s];
            shi[threadIdx.x] += shi[threadIdx.x + s];
            shq[threadIdx.x] += shq[threadIdx.x + s];
        }
        __syncthreads();
    }
    if (threadIdx.x == 0) {
        float n  = (float)(TP_ * cs);
        float mr = shr[0] / n, mi = shi[0] / n;
        float var = shq[0] / n - (mr*mr + mi*mi);   // E|x|^2 - |mu|^2
        stats[g*4+0] = mr; stats[g*4+1] = mi;
        stats[g*4+2] = rsqrtf(var + 1e-5f);
    }
}

__global__ void gn_apply_kernel(float2* __restrict__ x, const float* __restrict__ stats,
                                const float2* __restrict__ gamma, const float2* __restrict__ beta) {
    int tid = blockIdx.x * blockDim.x + threadIdx.x;
    if (tid >= TP_ * H_) return;
    int h = tid % H_;
    int g = h / (H_ / GROUPS_);
    float mr = stats[g*4+0], mi = stats[g*4+1], inv = stats[g*4+2];
    float2 v = x[tid];
    float nr = (v.x - mr) * inv, ni = (v.y - mi) * inv;
    float2 gm = gamma[h], bt = beta[h];
    x[tid] = make_float2(nr*gm.x - ni*gm.y + bt.x,
                         nr*gm.y + ni*gm.x + bt.y);
}

// ---------------------------------------------------------------------------
// 4) Complex GEMM: register-blocked WMMA + DOUBLE-BUFFERED async LDS staging.
//    Y[T,2304] = X[T,768] @ W[768,2304]  (complex64)
//
//    Block: 256 thr = 8 waves = 4 (M) x 2 (N); block tile 128x128.
//    Wave: 2 M-tiles x 4 N-tiles (32x64); Cr = sum aR*bR + sum (-aI)*bI
//    (f32 WMMA has no A/B NEG bits, only CNeg -> negate aI in VALU).
//    Software pipeline: while computing slab s from buffer p, the async
//    engine (ASYNCcnt) fills buffer p^1 with slab s+1; s_wait_asynccnt 0 +
//    barrier close the iteration. LDS: 2*(18K+16K)=68KB of the 320KB WGP pool.
//    As rows padded to 18 float2 (144B stride -> bank stride 36, conflict-free
//    over 16 lanes; keeps 16B alignment for b128 async writes).
//    OOB M rows clamped (not masked) so EXEC stays all-ones for WMMA.
// ---------------------------------------------------------------------------
#define BM_ 128
#define BN_ 128
#define KB_ 16

__global__ __launch_bounds__(256)
void cgemm_wmma_kernel(const float2* __restrict__ X, const float2* __restrict__ W,
                       float2* __restrict__ Y, int T) {
    constexpr int Kd = H_;         // 768
    constexpr int Nd = H3_;        // 2304
    constexpr int NS = Kd / KB_;   // 48 k-slabs
    __shared__ float2 As[2][BM_][KB_ + 2];   // 2 x 18 KB
    __shared__ float2 Bs[2][KB_][BN_];       // 2 x 16 KB

    const int gn   = Nd / BN_;            // 18
    const int bm   = blockIdx.x / gn;
    const int bn   = blockIdx.x % gn;
    const int tid  = threadIdx.x;
    const int wave = tid >> 5, lane = tid & 31;
    const int lm   = lane & 15, half = lane >> 4;
    const int wm   = wave & 3;            // M-wave 0..3
    const int wn   = wave >> 2;           // N-wave 0..1
    const int row0 = bm * BM_ + wm * 32;
    const int col0 = bn * BN_ + wn * 64;

    v8f accR[2][4] = {};
    v8f accI[2][4] = {};

    // cooperative stage of slab s into buffer p (8 x 16B chunks per thread)
    auto stage = [&](int s, int p) {
        const int k0 = s * KB_;
        #pragma unroll
        for (int it = 0; it < 4; ++it) {          // A: 128 rows x 16 complex
            int e  = tid + 256 * it;              // 0..1023
            int r  = e >> 3;
            int kp = (e & 7) * 2;
            int gr = bm * BM_ + r;
            if (gr > T - 1) gr = T - 1;           // clamp (garbage rows never stored)
            const float2* gp = X + (size_t)gr * Kd + (k0 + kp);
#if USE_ASYNC_LDS
            uint32_t lo = (uint32_t)(uintptr_t)&As[p][r][kp];
            asm volatile("global_load_async_to_lds_b128 %0, %1, off"
                         :: "v"(lo), "v"(gp) : "memory");
#else
            *(float4*)&As[p][r][kp] = *(const float4*)gp;
#endif
        }
        #pragma unroll
        for (int it = 0; it < 4; ++it) {          // B: 16 k-rows x 128 complex
            int e  = tid + 256 * it;
            int r  = e >> 6;
            int cp = (e & 63) * 2;
            const float2* gp = W + (size_t)(k0 + r) * Nd + bn * BN_ + cp;
#if USE_ASYNC_LDS
            uint32_t lo = (uint32_t)(uintptr_t)&Bs[p][r][cp];
            asm volatile("global_load_async_to_lds_b128 %0, %1, off"
                         :: "v"(lo), "v"(gp) : "memory");
#else
            *(float4*)&Bs[p][r][cp] = *(const float4*)gp;
#endif
        }
    };

    // prologue: fill buffer 0
    stage(0, 0);
    wait_async_all();
    __syncthreads();

    for (int s = 0; s < NS; ++s) {
        const int p = s & 1;
        // prefetch next slab into the other buffer while we compute this one
        if (s + 1 < NS) stage(s + 1, p ^ 1);

        // ---- compute: 4 k-steps of 4, 32 WMMAs each ----
        #pragma unroll
        for (int ks = 0; ks < 4; ++ks) {
            const int ka = ks * 4 + 2 * half;
            v2f aR[2], aI[2], aN[2];
            #pragma unroll
            for (int mt = 0; mt < 2; ++mt) {
                const int rl = wm * 32 + mt * 16 + lm;
                float2 a0 = As[p][rl][ka];
                float2 a1 = As[p][rl][ka + 1];
                aR[mt] = (v2f){ a0.x,  a1.x};
                aI[mt] = (v2f){ a0.y,  a1.y};
                aN[mt] = (v2f){-a0.y, -a1.y};
            }
            v2f bR[4], bI[4];
            #pragma unroll
            for (int nt = 0; nt < 4; ++nt) {
                const int cl = wn * 64 + nt * 16 + lm;
                float2 b0 = Bs[p][ka][cl];
                float2 b1 = Bs[p][ka + 1][cl];
                bR[nt] = (v2f){b0.x, b1.x};
                bI[nt] = (v2f){b0.y, b1.y};
            }
            #pragma unroll
            for (int mt = 0; mt < 2; ++mt) {
                #pragma unroll
                for (int nt = 0; nt < 4; ++nt) {
                    accR[mt][nt] = wmma_f32(aR[mt], bR[nt], accR[mt][nt]);
                    accR[mt][nt] = wmma_f32(aN[mt], bI[nt], accR[mt][nt]);
                    accI[mt][nt] = wmma_f32(aR[mt], bI[nt], accI[mt][nt]);
                    accI[mt][nt] = wmma_f32(aI[mt], bR[nt], accI[mt][nt]);
                }
            }
        }

        // close the pipeline stage: next-slab DMA done + all waves past reads
        wait_async_all();
        __syncthreads();
    }

    // ---- epilogue: C/D layout VGPR v -> M = v + 8*half, N = lane%16 ----
    #pragma unroll
    for (int mt = 0; mt < 2; ++mt) {
        #pragma unroll
        for (int nt = 0; nt < 4; ++nt) {
            const int n = col0 + nt * 16 + lm;
            #pragma unroll
            for (int v = 0; v < 8; ++v) {
                int m = row0 + mt * 16 + v + 8 * half;
                if (m < T)
                    Y[(size_t)m * Nd + n] =
                        make_float2(accR[mt][nt][v], accI[mt][nt][v]);
            }
        }
    }
}

// ---------------------------------------------------------------------------
// 5) GRU gating (elementwise, complex): hn = (1-z)*h + z*tanh(ax + r*ghh + b)
// ---------------------------------------------------------------------------
__global__ void gru_kernel(const float2* __restrict__ gx, const float2* __restrict__ gh,
                           const float2* __restrict__ h0, const float2* __restrict__ b,
                           float2* __restrict__ hn) {
    int tid = blockIdx.x * blockDim.x + threadIdx.x;
    if (tid >= TP_ * H_) return;
    int t = tid / H_, hh = tid % H_;
    const float2* gxr = gx + (size_t)t * H3_;
    const float2* ghr = gh + (size_t)t * H3_;

    float2 zin = make_float2(gxr[hh].x + ghr[hh].x + b[hh].x,
                             gxr[hh].y + ghr[hh].y + b[hh].y);
    float2 rin = make_float2(gxr[H_+hh].x + ghr[H_+hh].x + b[H_+hh].x,
                             gxr[H_+hh].y + ghr[H_+hh].y + b[H_+hh].y);
    float2 z = make_float2(sigm(zin.x), sigm(zin.y));
    float2 r = make_float2(sigm(rin.x), sigm(rin.y));

    float2 rg = cmul(r, ghr[2*H_+hh]);
    float2 ax = gxr[2*H_+hh];
    float2 bv = b[2*H_+hh];
    float2 a = make_float2(tanhf(ax.x + rg.x + bv.x), tanhf(ax.y + rg.y + bv.y));

    float2 h   = h0[tid];
    float2 omz = make_float2(1.f - z.x, -z.y);
    float2 t1  = cmul(omz, h);
    float2 t2  = cmul(z, a);
    hn[tid] = make_float2(t1.x + t2.x, t1.y + t2.y);
}

// ---------------------------------------------------------------------------
// 6) Transposed conv (stride 1, SAME) + trim + scale:
//    yfull[t,o] = sum_{j,h} hn[t+1-j, h] * dw[j,h,o];  y[f,:] = -LAM*(yfull[f+2,:]+db)
// ---------------------------------------------------------------------------
__global__ void deconv_kernel(const float2* __restrict__ hn, const float2* __restrict__ dw,
                              const float2* __restrict__ db, float* __restrict__ yout) {
    int tid = blockIdx.x * blockDim.x + threadIdx.x;
    if (tid >= F_ * OUT_) return;
    int f = tid / OUT_, o = tid % OUT_;
    int t = f + 2;                                   // trim = ceil((TP-F)/2) = 2
    float ar = 0.f, ai = 0.f;
    for (int j = 0; j < GS_; ++j) {
        int ts = t + 1 - j;                          // SAME transpose alignment
        if (ts < 0 || ts >= TP_) continue;
        const float2* hr = hn + (size_t)ts * H_;
        const float2* wr = dw + ((size_t)j * H_) * OUT_ + o;   // (GS,H,OUT)
        for (int h = 0; h < H_; ++h) {
            float2 xv = hr[h];
            float2 wv = wr[(size_t)h * OUT_];
            ar += xv.x*wv.x - xv.y*wv.y;
            ai += xv.x*wv.y + xv.y*wv.x;
        }
    }
    float2 bv = db[o];
    ar = -LAM_ * (ar + bv.x);
    ai = -LAM_ * (ai + bv.y);
    int a = o / M_, m = o % M_;
    size_t oi = (((size_t)a * F_ + f) * M_ + m) * 2; // (A,F,M) complex interleaved
    yout[oi]   = ar;
    yout[oi+1] = ai;
}

// ---------------------------------------------------------------------------
// Launcher
// ---------------------------------------------------------------------------
extern "C" void kernel_launch(void* const* d_in, const int* in_sizes, int n_in,
                              void* d_out, int out_size, void* d_ws, size_t ws_size,
                              hipStream_t stream) {
    const float2* fi    = (const float2*)d_in[0];   // (A,F,M,K) complex64
    const float2* u     = (const float2*)d_in[1];   // (F,P)
    const float2* h0    = (const float2*)d_in[2];   // (L,TP,H)
    const float2* cw    = (const float2*)d_in[3];   // (GS,CIN,H)
    const float2* cb    = (const float2*)d_in[4];   // (H,)
    const float2* gamma = (const float2*)d_in[5];   // (H,)
    const float2* beta  = (const float2*)d_in[6];   // (H,)
    const float2* wi    = (const float2*)d_in[7];   // (L,H,3H)
    const float2* wh    = (const float2*)d_in[8];   // (L,H,3H)
    const float2* bb    = (const float2*)d_in[9];   // (L,3H)
    const float2* dw    = (const float2*)d_in[10];  // (GS,H,OUT)
    const float2* db    = (const float2*)d_in[11];  // (OUT,)

    // Workspace carve-up (~178 MB total)
    char* ws = (char*)d_ws;
    size_t off = 0;
    auto carve = [&](size_t bytes) -> void* {
        void* p = ws + off;
        off = (off + bytes + 255) & ~(size_t)255;
        return p;
    };
    float2* xpad  = (float2*)carve((size_t)XROWS_ * CIN_ * sizeof(float2)); // ~1 MB
    float2* xact  = (float2*)carve((size_t)TP_ * H_  * sizeof(float2));     // ~25 MB
    float2* gx    = (float2*)carve((size_t)TP_ * H3_ * sizeof(float2));     // ~76 MB
    float2* gh    = (float2*)carve((size_t)TP_ * H3_ * sizeof(float2));     // ~76 MB
    float*  stats = (float*) carve(GROUPS_ * 4 * sizeof(float));

    float*  outf = (float*)d_out;
    // d_out = [ y (A,F,M) complex | hs (L,TP,H) complex ], interleaved floats.
    float2* hs = (float2*)(outf + (size_t)2 * A_ * F_ * M_);

    const int TPB = 256;

    // 1) pack + compress + pad
    {
        int n = XROWS_ * CIN_;
        prep_kernel<<<(n + TPB - 1) / TPB, TPB, 0, stream>>>(fi, u, xpad);
    }
    // 2) conv + bias + crelu
    {
        int n = TP_ * H_;
        conv_kernel<<<(n + TPB - 1) / TPB, TPB, 0, stream>>>(xpad, cw, cb, xact);
    }
    // 3) group norm
    gn_stats_kernel<<<GROUPS_, TPB, 0, stream>>>(xact, stats);
    {
        int n = TP_ * H_;
        gn_apply_kernel<<<(n + TPB - 1) / TPB, TPB, 0, stream>>>(xact, stats, gamma, beta);
    }
    // 4) two GRU layers: 2 complex WMMA GEMMs + elementwise gating each
    const int gm = (TP_ + BM_ - 1) / BM_;     // 33
    const int gn = H3_ / BN_;                 // 18
    const int gblocks = gm * gn;              // 594 blocks, 128x128 tile each
    const float2* xin = xact;
    for (int l = 0; l < L_; ++l) {
        const float2* wil = wi + (size_t)l * H_ * H3_;
        const float2* whl = wh + (size_t)l * H_ * H3_;
        const float2* h0l = h0 + (size_t)l * TP_ * H_;
        const float2* bl  = bb + (size_t)l * H3_;
        float2* hnl = hs + (size_t)l * TP_ * H_;

        cgemm_wmma_kernel<<<gblocks, TPB, 0, stream>>>(xin, wil, gx, TP_);
        cgemm_wmma_kernel<<<gblocks, TPB, 0, stream>>>(h0l, whl, gh, TP_);
        {
            int n = TP_ * H_;
            gru_kernel<<<(n + TPB - 1) / TPB, TPB, 0, stream>>>(gx, gh, h0l, bl, hnl);
        }
        xin = hnl;   // layer l output feeds layer l+1
    }
    // 5) transposed conv + trim + scale -> y region of d_out
    {
        int n = F_ * OUT_;
        deconv_kernel<<<(n + TPB - 1) / TPB, TPB, 0, stream>>>(
            hs + (size_t)(L_ - 1) * TP_ * H_, dw, db, outf);
    }
}